// TGCNCell_17334488007012
// MI455X (gfx1250) — compile-verified
//
#include <hip/hip_runtime.h>
#include <math.h>

// ---------------------------------------------------------------------------
// T-GCN cell for MI455X (gfx1250, wave32).
// GEMMs: V_WMMA_F32_16X16X4_F32 (exact fp32); tiles double-buffered in LDS via
// GLOBAL_LOAD_ASYNC_TO_LDS_B128 (ASYNCcnt) so copies overlap the matrix pipe.
// Dead fc-branch of the reference (_x1fc/_c1 unused) is skipped entirely.
// ---------------------------------------------------------------------------

#define NUNITS 64
#define NNODES 2048
#define CNODES 256
#define BATCH_ 64
#define FDIM   65               // INPUT_DIM + NUM_UNITS
#define FB     (FDIM * BATCH_)  // 4160
#define NU     (NNODES * NUNITS)

typedef __attribute__((ext_vector_type(2))) float v2f;
typedef __attribute__((ext_vector_type(8))) float v8f;

__device__ __forceinline__ float sigmoidf_(float x) {
  return 1.0f / (1.0f + __expf(-x));
}

// 16B global -> LDS via the CDNA5 async copy path (tracked by ASYNCcnt).
__device__ __forceinline__ void async_copy16(unsigned lds_off,
                                             const float* gptr) {
  asm volatile("global_load_async_to_lds_b128 %0, %1, off"
               :: "v"(lds_off), "v"((unsigned long long)(uintptr_t)gptr)
               : "memory");
}
template <int N>
__device__ __forceinline__ void async_wait() {
  asm volatile("s_wait_asynccnt %0" :: "i"(N) : "memory");
}

// ---------------------------------------------------------------------------
// x0[n, f*B+b] = f==0 ? inputs[b,n] : S[b, n*U + f-1]
// One node per block; 64x64 (f x b) transpose through LDS so state reads
// (coalesced along f) and x0 writes (coalesced along b) are both clean.
// ---------------------------------------------------------------------------
__global__ __launch_bounds__(256) void build_x0_kernel(
    const float* __restrict__ in, const float* __restrict__ S,
    float* __restrict__ x0)
{
  __shared__ float T[64][65];
  const int n = blockIdx.x;
  const int tid = threadIdx.x;
#pragma unroll
  for (int i = 0; i < 16; ++i) {
    int idx = tid + i * 256;            // 0..4095
    int b = idx >> 6, f = idx & 63;     // f fastest -> coalesced read
    T[b][f] = S[(size_t)b * NU + n * NUNITS + f];
  }
  __syncthreads();
  if (tid < 64)                          // f==0 feature row
    x0[(size_t)n * FB + tid] = in[tid * NNODES + n];
#pragma unroll
  for (int i = 0; i < 16; ++i) {
    int idx = tid + i * 256;
    int f = idx >> 6, b = idx & 63;     // b fastest -> coalesced write
    x0[(size_t)n * FB + (f + 1) * BATCH_ + b] = T[b][f];
  }
}

// ---------------------------------------------------------------------------
// D(MxN) = A(MxK) @ B(KxN), row-major. Block tile 128x64, 8 waves (256 thr),
// each wave a 32x32 subtile = 2x2 WMMA 16x16 f32 accumulators.
// Double-buffered async LDS staging: while buffer `cur` feeds the WMMAs, the
// next K-chunk streams into `cur^1`; s_wait_asynccnt<=3 retires only the
// current tile's 3 in-order copies.
// MODE 0: D = acc   MODE 1: D = sigmoid(acc)   MODE 2: D += acc
// Requires M%128==0, N%64==0, K%16==0 (true at all call sites).
// ---------------------------------------------------------------------------
template <int MODE>
__global__ __launch_bounds__(256) void gemm128_wmma(
    const float* __restrict__ A, const float* __restrict__ B,
    float* __restrict__ D, int M, int K, int N)
{
  __shared__ __attribute__((aligned(16))) float As[2][128][20]; // padded
  __shared__ __attribute__((aligned(16))) float Bs[2][16][68];

  const int tid   = threadIdx.x;
  const int wave  = tid >> 5, lane = tid & 31;
  const int lhalf = lane >> 4, l16 = lane & 15;
  const int waveM = wave & 3, waveN = wave >> 2;       // 4 x 2 wave grid
  const int mBlk  = blockIdx.x * 128, nBlk = blockIdx.y * 64;

  // per-thread staging coordinates (3 async b128 copies per thread per chunk)
  const int ra0 = tid >> 2,          ca0 = (tid & 3) << 2;
  const int ra1 = (tid + 256) >> 2,  ca1 = ((tid + 256) & 3) << 2;
  const int rb  = tid >> 4,          cb  = (tid & 15) << 2;

  auto stage = [&](int buf, int k0) {
    async_copy16((unsigned)(uintptr_t)&As[buf][ra0][ca0],
                 &A[(size_t)(mBlk + ra0) * K + k0 + ca0]);
    async_copy16((unsigned)(uintptr_t)&As[buf][ra1][ca1],
                 &A[(size_t)(mBlk + ra1) * K + k0 + ca1]);
    async_copy16((unsigned)(uintptr_t)&Bs[buf][rb][cb],
                 &B[(size_t)(k0 + rb) * N + nBlk + cb]);
  };

  v8f acc[2][2] = {};
  const int niter = K >> 4;

  stage(0, 0);                                   // prologue
  for (int it = 0; it < niter; ++it) {
    const int cur = it & 1;
    __syncthreads();                             // cur^1 no longer being read
    if (it + 1 < niter) {
      stage(cur ^ 1, (it + 1) << 4);             // prefetch next chunk
      async_wait<3>();                           // cur's 3 copies retired
    } else {
      async_wait<0>();
    }
    __syncthreads();                             // cur visible to all waves

#pragma unroll
    for (int kk = 0; kk < 16; kk += 4) {
      // A frag (16x4 f32): row = lane%16, k = 2*(lane/16)+j
      v2f a[2], bf[2];
#pragma unroll
      for (int mi = 0; mi < 2; ++mi) {
        int m = waveM * 32 + mi * 16 + l16;
        a[mi].x = As[cur][m][kk + 2 * lhalf + 0];
        a[mi].y = As[cur][m][kk + 2 * lhalf + 1];
      }
#pragma unroll
      for (int ni = 0; ni < 2; ++ni) {
        int n = waveN * 32 + ni * 16 + l16;
        bf[ni].x = Bs[cur][kk + 2 * lhalf + 0][n];
        bf[ni].y = Bs[cur][kk + 2 * lhalf + 1][n];
      }
#pragma unroll
      for (int mi = 0; mi < 2; ++mi)
#pragma unroll
        for (int ni = 0; ni < 2; ++ni)
          acc[mi][ni] = __builtin_amdgcn_wmma_f32_16x16x4_f32(
              false, a[mi], false, bf[ni], (short)0, acc[mi][ni],
              false, false);
    }
  }

  // C/D layout: VGPR r -> M = r + 8*(lane/16), col = lane%16
#pragma unroll
  for (int mi = 0; mi < 2; ++mi)
#pragma unroll
    for (int ni = 0; ni < 2; ++ni)
#pragma unroll
      for (int r = 0; r < 8; ++r) {
        int m = mBlk + waveM * 32 + mi * 16 + r + 8 * lhalf;
        int n = nBlk + waveN * 32 + ni * 16 + l16;
        size_t off = (size_t)m * N + n;
        float v = acc[mi][ni][r];
        if (MODE == 0)      D[off] = v;
        else if (MODE == 1) D[off] = sigmoidf_(v);
        else                D[off] += v;
      }
}

// ---------------------------------------------------------------------------
// Pass-1 projection + GRU gates. x1g[r=b*N+n, f] = x1[n, f*B+b].
// z = x1g @ w0 + b0; v = sigmoid(z); r_state = v[:,:64]*state; u_buf = v[:,64:]
// w0 staged zero-padded (68x128) in LDS -> unguarded ds reads; gathered x1
// fragments use clamped address + select (no exec divergence).
// ---------------------------------------------------------------------------
__global__ __launch_bounds__(256) void proj1_kernel(
    const float* __restrict__ x1, const float* __restrict__ w,
    const float* __restrict__ bias, const float* __restrict__ state,
    float* __restrict__ r_state, float* __restrict__ u_buf)
{
  __shared__ float Ws[68 * 128];
  __shared__ float BiasS[128];
  const int tid = threadIdx.x, wave = tid >> 5, lane = tid & 31;
  const int lhalf = lane >> 4, l16 = lane & 15;

  for (int idx = tid; idx < 68 * 128; idx += 256) {
    int f = idx >> 7, c = idx & 127;
    Ws[idx] = (f < FDIM) ? w[f * 128 + c] : 0.f;
  }
  if (tid < 128) BiasS[tid] = bias[tid];
  __syncthreads();

  const int rowBase = blockIdx.x * 128 + wave * 16;
  const int rA = rowBase + l16;
  const int bA = rA >> 11;                 // / NNODES
  const int nA = rA & (NNODES - 1);
  const float* x1r = x1 + (size_t)nA * FB + bA;

  v8f acc[8] = {};
  for (int kk = 0; kk < 68; kk += 4) {     // K padded 65 -> 68
    int f0  = kk + 2 * lhalf;
    int fa0 = (f0 < FDIM) ? f0 : 0;        // clamp: load always, select 0
    int fa1 = (f0 + 1 < FDIM) ? f0 + 1 : 0;
    float ax = x1r[fa0 * BATCH_];
    float ay = x1r[fa1 * BATCH_];
    v2f a;
    a.x = (f0     < FDIM) ? ax : 0.f;
    a.y = (f0 + 1 < FDIM) ? ay : 0.f;
#pragma unroll
    for (int ct = 0; ct < 8; ++ct) {
      int col = ct * 16 + l16;
      v2f bf;
      bf.x = Ws[(f0 + 0) * 128 + col];
      bf.y = Ws[(f0 + 1) * 128 + col];
      acc[ct] = __builtin_amdgcn_wmma_f32_16x16x4_f32(
          false, a, false, bf, (short)0, acc[ct], false, false);
    }
  }

#pragma unroll
  for (int ct = 0; ct < 8; ++ct)
#pragma unroll
    for (int r = 0; r < 8; ++r) {
      int m = rowBase + r + 8 * lhalf;
      int b = m >> 11, n = m & (NNODES - 1);
      int col = ct * 16 + l16;
      float v = sigmoidf_(acc[ct][r] + BiasS[col]);
      size_t so = (size_t)b * NU + (size_t)n * NUNITS;
      if (col < NUNITS) r_state[so + col] = v * state[so + col];
      else              u_buf[so + (col - NUNITS)] = v;
    }
}

// ---------------------------------------------------------------------------
// Pass-2 projection: c = tanh(x1g @ w1 + b1); out = u*state + (1-u)*c
// ---------------------------------------------------------------------------
__global__ __launch_bounds__(256) void proj2_kernel(
    const float* __restrict__ x1, const float* __restrict__ w,
    const float* __restrict__ bias, const float* __restrict__ state,
    const float* __restrict__ u_buf, float* __restrict__ out)
{
  __shared__ float Ws[68 * 64];
  __shared__ float BiasS[64];
  const int tid = threadIdx.x, wave = tid >> 5, lane = tid & 31;
  const int lhalf = lane >> 4, l16 = lane & 15;

  for (int idx = tid; idx < 68 * 64; idx += 256) {
    int f = idx >> 6, c = idx & 63;
    Ws[idx] = (f < FDIM) ? w[f * 64 + c] : 0.f;
  }
  if (tid < 64) BiasS[tid] = bias[tid];
  __syncthreads();

  const int rowBase = blockIdx.x * 128 + wave * 16;
  const int rA = rowBase + l16;
  const int bA = rA >> 11;
  const int nA = rA & (NNODES - 1);
  const float* x1r = x1 + (size_t)nA * FB + bA;

  v8f acc[4] = {};
  for (int kk = 0; kk < 68; kk += 4) {
    int f0  = kk + 2 * lhalf;
    int fa0 = (f0 < FDIM) ? f0 : 0;
    int fa1 = (f0 + 1 < FDIM) ? f0 + 1 : 0;
    float ax = x1r[fa0 * BATCH_];
    float ay = x1r[fa1 * BATCH_];
    v2f a;
    a.x = (f0     < FDIM) ? ax : 0.f;
    a.y = (f0 + 1 < FDIM) ? ay : 0.f;
#pragma unroll
    for (int ct = 0; ct < 4; ++ct) {
      int col = ct * 16 + l16;
      v2f bf;
      bf.x = Ws[(f0 + 0) * 64 + col];
      bf.y = Ws[(f0 + 1) * 64 + col];
      acc[ct] = __builtin_amdgcn_wmma_f32_16x16x4_f32(
          false, a, false, bf, (short)0, acc[ct], false, false);
    }
  }

#pragma unroll
  for (int ct = 0; ct < 4; ++ct)
#pragma unroll
    for (int r = 0; r < 8; ++r) {
      int m = rowBase + r + 8 * lhalf;
      int b = m >> 11, n = m & (NNODES - 1);
      int col = ct * 16 + l16;
      float c = tanhf(acc[ct][r] + BiasS[col]);
      size_t so = (size_t)b * NU + (size_t)n * NUNITS + col;
      float u = u_buf[so];
      out[so] = u * state[so] + (1.0f - u) * c;
    }
}

extern "C" void kernel_launch(void* const* d_in, const int* in_sizes, int n_in,
                              void* d_out, int out_size, void* d_ws, size_t ws_size,
                              hipStream_t stream)
{
  const float* inputs = (const float*)d_in[0];
  const float* state  = (const float*)d_in[1];
  const float* adj    = (const float*)d_in[2];
  // d_in[3] = adj1 (dead branch)
  const float* afc    = (const float*)d_in[4];
  const float* afct   = (const float*)d_in[5];
  const float* w0     = (const float*)d_in[6];
  const float* b0     = (const float*)d_in[7];
  const float* w1     = (const float*)d_in[8];
  const float* b1     = (const float*)d_in[9];
  // d_in[10..13] = w01,b01,w11,b11 (dead branch)

  float* x0     = (float*)d_ws;                      // N  x FB
  float* x0fc   = x0     + (size_t)NNODES * FB;      // C  x FB
  float* x1     = x0fc   + (size_t)CNODES * FB;      // N  x FB
  float* rstate = x1     + (size_t)NNODES * FB;      // B  x N*U
  float* ubuf   = rstate + (size_t)BATCH_ * NU;      // B  x N*U

  dim3 blk(256);
  dim3 gC(CNODES / 128, FB / 64);                    // (2, 65)
  dim3 gN(NNODES / 128, FB / 64);                    // (16, 65)
  int prBlocks = (BATCH_ * NNODES) / 128;            // 1024

  // ---- pass 1 (gate GEMMs) ----
  build_x0_kernel<<<NNODES, blk, 0, stream>>>(inputs, state, x0);
  gemm128_wmma<0><<<gC, blk, 0, stream>>>(afc,  x0,   x0fc, CNODES, NNODES, FB);
  gemm128_wmma<1><<<gN, blk, 0, stream>>>(afct, x0fc, x1,   NNODES, CNODES, FB);
  gemm128_wmma<2><<<gN, blk, 0, stream>>>(adj,  x0,   x1,   NNODES, NNODES, FB);
  proj1_kernel<<<prBlocks, blk, 0, stream>>>(x1, w0, b0, state, rstate, ubuf);

  // ---- pass 2 (candidate GEMMs, state <- r*state) ----
  build_x0_kernel<<<NNODES, blk, 0, stream>>>(inputs, rstate, x0);
  gemm128_wmma<0><<<gC, blk, 0, stream>>>(afc,  x0,   x0fc, CNODES, NNODES, FB);
  gemm128_wmma<1><<<gN, blk, 0, stream>>>(afct, x0fc, x1,   NNODES, CNODES, FB);
  gemm128_wmma<2><<<gN, blk, 0, stream>>>(adj,  x0,   x1,   NNODES, NNODES, FB);
  proj2_kernel<<<prBlocks, blk, 0, stream>>>(x1, w1, b1, state, ubuf,
                                             (float*)d_out);
}